// GNN_4870492914241
// MI455X (gfx1250) — compile-verified
//
#include <hip/hip_runtime.h>

// ============================================================================
// Compile-time Clebsch-Gordan tensors (mirrors the Python reference exactly,
// evaluated in constexpr double precision, normalized by Frobenius norm).
// ============================================================================
namespace cgc {

constexpr double fact(int n) {
  double r = 1.0;
  for (int i = 2; i <= n; ++i) r *= (double)i;
  return r;
}

constexpr double csqrt(double x) {
  if (x <= 0.0) return 0.0;
  double r = x > 1.0 ? x : 1.0;
  for (int i = 0; i < 80; ++i) r = 0.5 * (r + x / r);
  return r;
}

struct Cplx { double re; double im; };
constexpr Cplx cmul(Cplx a, Cplx b) { return { a.re*b.re - a.im*b.im, a.re*b.im + a.im*b.re }; }
constexpr Cplx cconj(Cplx a) { return { a.re, -a.im }; }
constexpr Cplx cadd(Cplx a, Cplx b) { return { a.re + b.re, a.im + b.im }; }
constexpr Cplx cscale(double s, Cplx a) { return { s*a.re, s*a.im }; }

constexpr double su2_cg_coef(int j1, int m1, int j2, int m2, int j3, int m3) {
  double pref = csqrt((2.0*j3 + 1.0) * fact(j3 + j1 - j2) * fact(j3 - j1 + j2) *
                      fact(j1 + j2 - j3) / fact(j1 + j2 + j3 + 1));
  pref *= csqrt(fact(j3 + m3) * fact(j3 - m3) * fact(j1 - m1) * fact(j1 + m1) *
                fact(j2 - m2) * fact(j2 + m2));
  double s = 0.0;
  for (int k = 0; k <= j1 + j2 - j3; ++k) {
    int d0 = k, d1 = j1 + j2 - j3 - k, d2 = j1 - m1 - k, d3 = j2 + m2 - k,
        d4 = j3 - j2 + m1 + k, d5 = j3 - j1 - m2 + k;
    if (d0 < 0 || d1 < 0 || d2 < 0 || d3 < 0 || d4 < 0 || d5 < 0) continue;
    double denom = fact(d0) * fact(d1) * fact(d2) * fact(d3) * fact(d4) * fact(d5);
    s += ((k & 1) ? -1.0 : 1.0) / denom;
  }
  return pref * s;
}

struct T3 { double c[5][5][5]; };   // [2l1+1][2l2+1][2l3+1], zero-padded
struct QM { Cplx q[5][5]; };

constexpr T3 su2_cg(int j1, int j2, int j3) {
  T3 C{};
  for (int m1 = -j1; m1 <= j1; ++m1)
    for (int m2 = -j2; m2 <= j2; ++m2) {
      int m3 = m1 + m2;
      if (m3 >= -j3 && m3 <= j3)
        C.c[j1 + m1][j2 + m2][j3 + m3] = su2_cg_coef(j1, m1, j2, m2, j3, m3);
    }
  return C;
}

constexpr QM real_to_complex(int l) {
  QM Q{};
  const double is2 = 1.0 / csqrt(2.0);
  for (int m = -l; m <= -1; ++m) {
    Q.q[l + m][l - m] = Cplx{ is2, 0.0 };
    Q.q[l + m][l + m] = Cplx{ 0.0, -is2 };
  }
  Q.q[l][l] = Cplx{ 1.0, 0.0 };
  for (int m = 1; m <= l; ++m) {
    double sgn = (m & 1) ? -1.0 : 1.0;
    Q.q[l + m][l + m] = Cplx{ sgn * is2, 0.0 };
    Q.q[l + m][l - m] = Cplx{ 0.0, sgn * is2 };
  }
  Cplx f = (l == 0) ? Cplx{ 1.0, 0.0 } : (l == 1) ? Cplx{ 0.0, -1.0 } : Cplx{ -1.0, 0.0 };
  for (int a = 0; a < 2*l + 1; ++a)
    for (int b = 0; b < 2*l + 1; ++b)
      Q.q[a][b] = cmul(f, Q.q[a][b]);
  return Q;
}

constexpr T3 so3_cg(int l1, int l2, int l3) {
  QM Q1 = real_to_complex(l1);
  QM Q2 = real_to_complex(l2);
  QM Q3 = real_to_complex(l3);
  T3 Csu = su2_cg(l1, l2, l3);
  T3 R{};
  const int n1 = 2*l1 + 1, n2 = 2*l2 + 1, n3 = 2*l3 + 1;
  double norm2 = 0.0;
  for (int j = 0; j < n1; ++j)
    for (int l = 0; l < n2; ++l)
      for (int m = 0; m < n3; ++m) {
        Cplx s{ 0.0, 0.0 };
        for (int i = 0; i < n1; ++i)
          for (int k = 0; k < n2; ++k)
            for (int n = 0; n < n3; ++n) {
              double cv = Csu.c[i][k][n];
              if (cv == 0.0) continue;
              Cplx t = cmul(Q1.q[i][j], Q2.q[k][l]);
              t = cmul(t, cconj(Q3.q[n][m]));
              s = cadd(s, cscale(cv, t));
            }
        R.c[j][l][m] = s.re;
        norm2 += s.re * s.re;
      }
  double nrm = csqrt(norm2);
  if (nrm > 0.0) {
    for (int j = 0; j < n1; ++j)
      for (int l = 0; l < n2; ++l)
        for (int m = 0; m < n3; ++m)
          R.c[j][l][m] /= nrm;
  }
  return R;
}

} // namespace cgc

constexpr cgc::T3 C000 = cgc::so3_cg(0, 0, 0);
constexpr cgc::T3 C110 = cgc::so3_cg(1, 1, 0);
constexpr cgc::T3 C220 = cgc::so3_cg(2, 2, 0);
constexpr cgc::T3 C011 = cgc::so3_cg(0, 1, 1);
constexpr cgc::T3 C101 = cgc::so3_cg(1, 0, 1);
constexpr cgc::T3 C111 = cgc::so3_cg(1, 1, 1);
constexpr cgc::T3 C121 = cgc::so3_cg(1, 2, 1);
constexpr cgc::T3 C211 = cgc::so3_cg(2, 1, 1);
constexpr cgc::T3 C221 = cgc::so3_cg(2, 2, 1);
constexpr cgc::T3 C022 = cgc::so3_cg(0, 2, 2);
constexpr cgc::T3 C112 = cgc::so3_cg(1, 1, 2);
constexpr cgc::T3 C122 = cgc::so3_cg(1, 2, 2);
constexpr cgc::T3 C202 = cgc::so3_cg(2, 0, 2);
constexpr cgc::T3 C212 = cgc::so3_cg(2, 1, 2);
constexpr cgc::T3 C222 = cgc::so3_cg(2, 2, 2);

static_assert(C000.c[0][0][0] == 1.0, "C000 must be exactly 1");
static_assert(C110.c[0][0][0] * C110.c[0][0][0] > 0.30 &&
              C110.c[0][0][0] * C110.c[0][0][0] < 0.37, "C110 diag ~ 1/sqrt(3)");
static_assert(C110.c[0][1][0] == 0.0, "C110 off-diagonal must be 0");
static_assert(C220.c[0][0][0] * C220.c[0][0][0] > 0.18 &&
              C220.c[0][0][0] * C220.c[0][0][0] < 0.22, "C220 diag ~ 1/sqrt(5)");

// ============================================================================
// Fused dense tensor: M2[k][i][j] = coef (CG) and cid (which runtime weight).
// ============================================================================
struct Fused {
  float coef[9][9][9];          // [k][i][j]
  unsigned char cid[9][9][9];   // index into wflat[15]
};

constexpr Fused make_fused() {
  Fused F{};
  struct Combo { int la, lh, lo, cid; const cgc::T3* t; };
  const int off[3] = { 0, 1, 4 };
  const Combo combos[15] = {
    {0,0,0, 0, &C000}, {1,1,0, 1, &C110}, {2,2,0, 2, &C220},
    {0,1,1, 3, &C011}, {1,0,1, 4, &C101}, {1,1,1, 5, &C111},
    {1,2,1, 6, &C121}, {2,1,1, 7, &C211}, {2,2,1, 8, &C221},
    {0,2,2, 9, &C022}, {1,1,2,10, &C112}, {1,2,2,11, &C122},
    {2,0,2,12, &C202}, {2,1,2,13, &C212}, {2,2,2,14, &C222},
  };
  for (int cb = 0; cb < 15; ++cb) {
    const Combo& c = combos[cb];
    const int na = 2*c.la + 1, nh = 2*c.lh + 1, no = 2*c.lo + 1;
    for (int i = 0; i < na; ++i)
      for (int j = 0; j < nh; ++j)
        for (int k = 0; k < no; ++k) {
          double v = c.t->c[i][j][k];
          if (v != 0.0) {
            F.coef[off[c.lo] + k][off[c.la] + i][off[c.lh] + j] = (float)v;
            F.cid [off[c.lo] + k][off[c.la] + i][off[c.lh] + j] = (unsigned char)c.cid;
          }
        }
  }
  return F;
}

__constant__ Fused FUSED = make_fused();

// ============================================================================
// Quad tiling of the 9x9 (i,j) product grid: 25 quads {2r,2r+1}x{2c,2c+1}
// (clamped at 8). Quad q = r*5 + c -> one V_WMMA_F32_16X16X4_F32, K slots:
//   K=0:(ia,j0)  K=1:(ia,j1)  K=2:(ib,j0)  K=3:(ib,j1)
// A operand (constants, 16k x 4K): lane m=L%16 is the k-row; half g=L/16 and
// register r give K=2g+r -> value M2[m][i(g)][j(r)] (zeroed for clamp dups).
// B operand (per-edge, 4K x 16e): lane e=L%16, K=2g+r ->
//   b[r] = (g ? A[ib] : A[ia]) * H[jr]
// D (16x16): lane e=L%16 holds k = v + 8g in VGPR v.
// ============================================================================
#define NQUAD 25

// ---- Prologue: fold runtime weights into the 25 pre-swizzled A-operand tiles.
__global__ void __launch_bounds__(256)
cg_fuse_weights(const float* __restrict__ w0, const float* __restrict__ w1,
                const float* __restrict__ w2, float* __restrict__ aop) {
  const int t = blockIdx.x * blockDim.x + threadIdx.x;
  if (t >= NQUAD * 32) return;
  const int q = t >> 5, lane = t & 31;
  const int m = lane & 15, g = lane >> 4;
  const int r = q / 5, c = q % 5;
  const int ia = 2 * r, ib = (2 * r + 1 > 8) ? 8 : 2 * r + 1;
  const int j0 = 2 * c, j1 = (2 * c + 1 > 8) ? 8 : 2 * c + 1;
  const int i = g ? ib : ia;
  const bool dup_i = (g == 1) && (ib == ia);

  float wflat[15];
  for (int x = 0; x < 3; ++x) wflat[x]     = w0[x];
  for (int x = 0; x < 6; ++x) wflat[3 + x] = w1[x];
  for (int x = 0; x < 6; ++x) wflat[9 + x] = w2[x];

  float v0 = 0.0f, v1 = 0.0f;
  if (m < 9 && !dup_i) {
    v0 = FUSED.coef[m][i][j0] * wflat[FUSED.cid[m][i][j0]];
    if (j1 != j0)
      v1 = FUSED.coef[m][i][j1] * wflat[FUSED.cid[m][i][j1]];
  }
  aop[t * 2 + 0] = v0;
  aop[t * 2 + 1] = v1;
}

// ---- Main kernel: streaming WMMA contraction, 16 edges per wave-tile.
typedef __attribute__((ext_vector_type(2))) float v2f;
typedef __attribute__((ext_vector_type(8))) float v8f;

__device__ __forceinline__ float nan_to_num_f(float x) {
  const float FMAX = 3.4028234663852886e+38f;
  if (__builtin_isnan(x)) return 0.0f;
  x = fmaxf(x, -FMAX);
  x = fminf(x,  FMAX);
  return x;
}

// One quad: b = {AB*Hj0, AB*Hj1}; D accumulate. All operands are named
// scalars so no private-array dynamic indexing can be introduced.
#define DO_QUAD(Q, AB, HJ0, HJ1, DST)                                        \
  {                                                                          \
    v2f b_;                                                                  \
    b_.x = (AB) * (HJ0);                                                     \
    b_.y = (AB) * (HJ1);                                                     \
    DST = __builtin_amdgcn_wmma_f32_16x16x4_f32(false, aop[Q], false, b_,    \
                                                (short)0, DST, false, false);\
  }

__global__ void __launch_bounds__(256)
cg_tp_wmma(const float* __restrict__ a0, const float* __restrict__ a1,
           const float* __restrict__ a2, const float* __restrict__ h0,
           const float* __restrict__ h1, const float* __restrict__ h2,
           const float* __restrict__ aop_tab, float* __restrict__ out,
           int E, int ntiles) {
  const int lane = threadIdx.x & 31;
  const int n = lane & 15;            // edge column within tile
  const bool ghi = (lane >> 4) != 0;  // half-wave (K-slot group)
  const int waveId = blockIdx.x * (blockDim.x >> 5) + (threadIdx.x >> 5);
  const int nwaves = gridDim.x * (blockDim.x >> 5);

  // Preload all 25 pre-swizzled constant A-operand tiles (50 VGPRs).
  v2f aop[NQUAD];
#pragma unroll
  for (int q = 0; q < NQUAD; ++q)
    aop[q] = *(const v2f*)(aop_tab + (q * 32 + lane) * 2);

  for (int tile = waveId; tile < ntiles; tile += nwaves) {
    const int base = tile * 16;
    int e = base + n;
    if (e >= E) e = E - 1;            // clamp loads; stores predicated below

    // Named scalars only (keeps every select a single v_cndmask).
    const float As0 = a0[e];
    const float Hs0 = h0[e];
    const float As1 = a1[e * 3 + 0], As2 = a1[e * 3 + 1], As3 = a1[e * 3 + 2];
    const float Hs1 = h1[e * 3 + 0], Hs2 = h1[e * 3 + 1], Hs3 = h1[e * 3 + 2];
    const float As4 = a2[e * 5 + 0], As5 = a2[e * 5 + 1], As6 = a2[e * 5 + 2];
    const float As7 = a2[e * 5 + 3], As8 = a2[e * 5 + 4];
    const float Hs4 = h2[e * 5 + 0], Hs5 = h2[e * 5 + 1], Hs6 = h2[e * 5 + 2];
    const float Hs7 = h2[e * 5 + 3], Hs8 = h2[e * 5 + 4];

    // Half-wave row blends: one v_cndmask each (AB4 folds to As8).
    const float AB0 = ghi ? As1 : As0;
    const float AB1 = ghi ? As3 : As2;
    const float AB2 = ghi ? As5 : As4;
    const float AB3 = ghi ? As7 : As6;
    const float AB4 = As8;

    v8f d0 = {};   // two independent accumulators to shorten the D->C chain
    v8f d1 = {};

    // r = 0 (rows {0,1})
    DO_QUAD( 0, AB0, Hs0, Hs1, d0);
    DO_QUAD( 1, AB0, Hs2, Hs3, d1);
    DO_QUAD( 2, AB0, Hs4, Hs5, d0);
    DO_QUAD( 3, AB0, Hs6, Hs7, d1);
    DO_QUAD( 4, AB0, Hs8, Hs8, d0);
    // r = 1 (rows {2,3})
    DO_QUAD( 5, AB1, Hs0, Hs1, d1);
    DO_QUAD( 6, AB1, Hs2, Hs3, d0);
    DO_QUAD( 7, AB1, Hs4, Hs5, d1);
    DO_QUAD( 8, AB1, Hs6, Hs7, d0);
    DO_QUAD( 9, AB1, Hs8, Hs8, d1);
    // r = 2 (rows {4,5})
    DO_QUAD(10, AB2, Hs0, Hs1, d0);
    DO_QUAD(11, AB2, Hs2, Hs3, d1);
    DO_QUAD(12, AB2, Hs4, Hs5, d0);
    DO_QUAD(13, AB2, Hs6, Hs7, d1);
    DO_QUAD(14, AB2, Hs8, Hs8, d0);
    // r = 3 (rows {6,7})
    DO_QUAD(15, AB3, Hs0, Hs1, d1);
    DO_QUAD(16, AB3, Hs2, Hs3, d0);
    DO_QUAD(17, AB3, Hs4, Hs5, d1);
    DO_QUAD(18, AB3, Hs6, Hs7, d0);
    DO_QUAD(19, AB3, Hs8, Hs8, d1);
    // r = 4 (row {8})
    DO_QUAD(20, AB4, Hs0, Hs1, d0);
    DO_QUAD(21, AB4, Hs2, Hs3, d1);
    DO_QUAD(22, AB4, Hs4, Hs5, d0);
    DO_QUAD(23, AB4, Hs6, Hs7, d1);
    DO_QUAD(24, AB4, Hs8, Hs8, d0);

    // D[k][e]: lane holds edge n; VGPR v holds k = v + 8g.
    const bool valid = (base + n) < E;
    if (valid) {
      float* op = out + (size_t)(base + n) * 9;
      if (!ghi) {
#pragma unroll
        for (int v = 0; v < 8; ++v) op[v] = nan_to_num_f(d0[v] + d1[v]);
      } else {
        op[8] = nan_to_num_f(d0[0] + d1[0]);
      }
    }
  }
}

extern "C" void kernel_launch(void* const* d_in, const int* in_sizes, int n_in,
                              void* d_out, int out_size, void* d_ws, size_t ws_size,
                              hipStream_t stream) {
  (void)n_in; (void)ws_size; (void)out_size;
  const float* a0 = (const float*)d_in[0];
  const float* a1 = (const float*)d_in[1];
  const float* a2 = (const float*)d_in[2];
  const float* h0 = (const float*)d_in[3];
  const float* h1 = (const float*)d_in[4];
  const float* h2 = (const float*)d_in[5];
  const float* w0 = (const float*)d_in[6];
  const float* w1 = (const float*)d_in[7];
  const float* w2 = (const float*)d_in[8];
  const int E = in_sizes[0];
  const int ntiles = (E + 15) / 16;

  float* aop = (float*)d_ws;   // NQUAD*32*2 floats = 6.4 KB

  hipLaunchKernelGGL(cg_fuse_weights, dim3((NQUAD * 32 + 255) / 256), dim3(256),
                     0, stream, w0, w1, w2, aop);

  hipLaunchKernelGGL(cg_tp_wmma, dim3(1024), dim3(256), 0, stream,
                     a0, a1, a2, h0, h1, h2, aop, (float*)d_out, E, ntiles);
}